// PoseAug_79800492360323
// MI455X (gfx1250) — compile-verified
//
#include <hip/hip_runtime.h>
#include <stdint.h>

// PoseAug for MI455X (gfx1250): bandwidth-bound (~1 FLOP/byte, ~420 MB total
// traffic -> ~18us floor @ 23.3 TB/s). No WMMA-shaped math exists here
// (batched 3x3 only -> block-diagonal, unmappable to 16x16 WMMA without
// paying more in layout shuffles than the 27 FMAs it replaces). The CDNA5-
// specific lever is the async global->LDS path: each wave stages its
// 32-batch tile of poses_3d+axis with global_load_async_to_lds_b128
// (ASYNCcnt), overlaps the copy with camera math, then s_wait_asynccnt.
// Round 2: replaced IEEE divisions/sqrt with raw v_rcp_f32/v_sqrt_f32 to
// cut VALU (~2K instr of div/sqrt fixup code) below the HBM roofline.

#define NJ  24
#define TPB 256                      // 8 waves of 32 (wave32)
#define WAVES_PER_BLOCK (TPB / 32)
#define TILE_FLOATS (32 * 72)        // 32 batches x 72 floats (poses or axis)

__device__ __forceinline__ float fsqrt_hw(float x) { return __builtin_amdgcn_sqrtf(x); }
__device__ __forceinline__ float frcp_hw(float x)  { return __builtin_amdgcn_rcpf(x); }

__global__ __launch_bounds__(TPB) void poseaug_kernel(
    const float* __restrict__ poses_3d, const float* __restrict__ rot_mats,
    const float* __restrict__ axis_in,  const float* __restrict__ angle_u,
    const float* __restrict__ scale_u,  const float* __restrict__ Kin,
    const float* __restrict__ Rin,      const float* __restrict__ tin,
    const float* __restrict__ focal_u,  const float* __restrict__ yaw_u,
    const float* __restrict__ pitch_u,  const float* __restrict__ t_jit,
    float* __restrict__ o_poses, float* __restrict__ o_2d,
    float* __restrict__ o_rots,  float* __restrict__ o_K,
    float* __restrict__ o_R,     float* __restrict__ o_t, int B)
{
    constexpr int PAR[NJ] = {-1,0,0,0,1,2,3,4,5,6,7,8,9,9,9,12,13,14,16,17,18,19,20,21};
    extern __shared__ float smem[];

    const int tid  = threadIdx.x;
    const int lane = tid & 31;
    const int wave = tid >> 5;
    const int b    = blockIdx.x * TPB + tid;
    const bool valid = (b < B);
    const int bC = valid ? b : (B - 1);

    // Wave-private tile base (clamped so the async copy never reads OOB).
    const int wb0raw = blockIdx.x * TPB + wave * 32;
    const int wb0    = (wb0raw + 32 <= B) ? wb0raw : (B - 32);
    const int li     = bC - wb0;            // 0..31, my slot in the tile

    float* wposes = smem + wave * (2 * TILE_FLOATS);
    float* waxis  = wposes + TILE_FLOATS;

    // ---------------- async stage: 2 x 9216 B per wave, coalesced b128 ----
    {
        const float* gp = poses_3d + (size_t)wb0 * 72;
        const float* ga = axis_in  + (size_t)wb0 * 72;
        const unsigned lp0 = (unsigned)(uintptr_t)wposes;  // low 32b = LDS offset
        const unsigned la0 = (unsigned)(uintptr_t)waxis;
        #pragma unroll
        for (int c = 0; c < 18; ++c) {
            unsigned go = (unsigned)(lane * 16) + (unsigned)c * 512u;
            asm volatile("global_load_async_to_lds_b128 %0, %1, %2"
                         :: "v"(lp0 + go), "v"(go), "s"(gp) : "memory");
        }
        #pragma unroll
        for (int c = 0; c < 18; ++c) {
            unsigned go = (unsigned)(lane * 16) + (unsigned)c * 512u;
            asm volatile("global_load_async_to_lds_b128 %0, %1, %2"
                         :: "v"(la0 + go), "v"(go), "s"(ga) : "memory");
        }
    }

    // ---------------- camera augmentation (overlaps the async copy) -------
    float kk[9], ra[9], ta[3];
    {
        const float* Kp = Kin + (size_t)bC * 9;
        const float* Rp = Rin + (size_t)bC * 9;
        float rr[9];
        #pragma unroll
        for (int q = 0; q < 9; ++q) kk[q] = Kp[q];
        #pragma unroll
        for (int q = 0; q < 9; ++q) rr[q] = Rp[q];
        float f0 = focal_u[(size_t)bC * 2 + 0], f1 = focal_u[(size_t)bC * 2 + 1];
        kk[0] *= 1.0f + (f0 * 2.0f - 1.0f) * 0.01f;   // CAM_JIT*0.1
        kk[4] *= 1.0f + (f1 * 2.0f - 1.0f) * 0.01f;
        float yw = (yaw_u[bC]   * 2.0f - 1.0f) * 0.01f;
        float pt = (pitch_u[bC] * 2.0f - 1.0f) * 0.005f;
        float cy = __cosf(yw), sy = __sinf(yw);
        float cp = __cosf(pt), sp = __sinf(pt);
        // Ryp = R_yaw @ R_pitch
        float m00 = cy,  m01 = sy * sp, m02 = sy * cp;
        float           m11 = cp,      m12 = -sp;
        float m20 = -sy, m21 = cy * sp, m22 = cy * cp;
        ra[0] = rr[0]*m00 + rr[2]*m20;
        ra[1] = rr[0]*m01 + rr[1]*m11 + rr[2]*m21;
        ra[2] = rr[0]*m02 + rr[1]*m12 + rr[2]*m22;
        ra[3] = rr[3]*m00 + rr[5]*m20;
        ra[4] = rr[3]*m01 + rr[4]*m11 + rr[5]*m21;
        ra[5] = rr[3]*m02 + rr[4]*m12 + rr[5]*m22;
        ra[6] = rr[6]*m00 + rr[8]*m20;
        ra[7] = rr[6]*m01 + rr[7]*m11 + rr[8]*m21;
        ra[8] = rr[6]*m02 + rr[7]*m12 + rr[8]*m22;
        #pragma unroll
        for (int q = 0; q < 3; ++q)
            ta[q] = tin[(size_t)bC * 3 + q] + t_jit[(size_t)bC * 3 + q] * 0.005f;
        if (valid) {
            #pragma unroll
            for (int q = 0; q < 9; ++q) o_K[(size_t)b * 9 + q] = kk[q];
            #pragma unroll
            for (int q = 0; q < 9; ++q) o_R[(size_t)b * 9 + q] = ra[q];
            #pragma unroll
            for (int q = 0; q < 3; ++q) o_t[(size_t)b * 3 + q] = ta[q];
        }
    }

    const float* angp = angle_u + (size_t)bC * NJ;
    const float* sclp = scale_u + (size_t)bC * NJ;
    const float* rmp  = rot_mats + (size_t)bC * (NJ * 9);
    float* orp = o_rots  + (size_t)bC * (NJ * 9);
    float* opp = o_poses + (size_t)bC * (NJ * 3);
    float* o2p = o_2d    + (size_t)bC * (NJ * 2);

    // ---------------- wait for the wave's tile (per-wave counter) ---------
    asm volatile("s_wait_asynccnt 0" ::: "memory");

    const float* myP = wposes + li * 72;
    const float* myA = waxis  + li * 72;

    float Px[NJ], Py[NJ], Pz[NJ];

    auto project_store = [&](int i, float x, float y, float z) {
        float cx = ra[0]*x + ra[1]*y + ra[2]*z + ta[0];
        float cv = ra[3]*x + ra[4]*y + ra[5]*z + ta[1];
        float cz = ra[6]*x + ra[7]*y + ra[8]*z + ta[2];
        float invz = frcp_hw(fmaxf(cz, 1e-5f));
        float u = kk[0] * (cx * invz) + kk[2];
        float v = kk[4] * (cv * invz) + kk[5];
        if (valid) {
            o2p[i*2+0] = u * (1.0f/256.0f) - 1.0f;   // /512*2 - 1
            o2p[i*2+1] = v * (1.0f/256.0f) - 1.0f;
        }
    };

    // joint 0: passthrough
    Px[0] = myP[0]; Py[0] = myP[1]; Pz[0] = myP[2];
    if (valid) {
        opp[0] = Px[0]; opp[1] = Py[0]; opp[2] = Pz[0];
        #pragma unroll
        for (int q = 0; q < 9; ++q) orp[q] = rmp[q];
    }
    project_store(0, Px[0], Py[0], Pz[0]);

    #pragma unroll
    for (int i = 1; i < NJ; ++i) {
        const int p = PAR[i];
        float ox = myP[i*3+0], oy = myP[i*3+1], oz = myP[i*3+2];
        float bx = ox - Px[p], by = oy - Py[p], bz = oz - Pz[p];
        float blen = fsqrt_hw(bx*bx + by*by + bz*bz);
        float ax = myA[i*3+0], ay = myA[i*3+1], az = myA[i*3+2];
        float inv1 = frcp_hw(fsqrt_hw(ax*ax + ay*ay + az*az) + 1e-8f);
        ax *= inv1; ay *= inv1; az *= inv1;
        float inv2 = frcp_hw(fsqrt_hw(ax*ax + ay*ay + az*az) + 1e-8f); // ref normalizes twice
        ax *= inv2; ay *= inv2; az *= inv2;
        float ang = (angp[i] * 2.0f - 1.0f) * 0.2f;
        float c1 = __cosf(ang), s1 = __sinf(ang);
        float k1 = 1.0f - c1;
        // rb = Ra @ bone = c*b + s*(a x b) + (1-c)(a.b)a
        float adb = ax*bx + ay*by + az*bz;
        float crx = ay*bz - az*by, cry = az*bx - ax*bz, crz = ax*by - ay*bx;
        float rbx = c1*bx + s1*crx + k1*adb*ax;
        float rby = c1*by + s1*cry + k1*adb*ay;
        float rbz = c1*bz + s1*crz + k1*adb*az;
        float invrn = frcp_hw(fsqrt_hw(rbx*rbx + rby*rby + rbz*rbz) + 1e-8f);
        float sc = 1.0f + (sclp[i] * 2.0f - 1.0f) * 0.1f;
        float f = sc * blen * invrn;
        Px[i] = Px[p] + rbx * f;
        Py[i] = Py[p] + rby * f;
        Pz[i] = Pz[p] + rbz * f;
        if (valid) { opp[i*3+0] = Px[i]; opp[i*3+1] = Py[i]; opp[i*3+2] = Pz[i]; }
        project_store(i, Px[i], Py[i], Pz[i]);

        // rots[i] = rot_mats[i] @ Ra   (Ra = c*I + s*K + (1-c) a a^T)
        float Ra00 = c1 + k1*ax*ax,     Ra01 = -s1*az + k1*ax*ay, Ra02 =  s1*ay + k1*ax*az;
        float Ra10 =  s1*az + k1*ay*ax, Ra11 = c1 + k1*ay*ay,     Ra12 = -s1*ax + k1*ay*az;
        float Ra20 = -s1*ay + k1*az*ax, Ra21 =  s1*ax + k1*az*ay, Ra22 = c1 + k1*az*az;
        if (valid) {
            #pragma unroll
            for (int r = 0; r < 3; ++r) {
                float m0 = rmp[i*9 + r*3 + 0];
                float m1 = rmp[i*9 + r*3 + 1];
                float m2 = rmp[i*9 + r*3 + 2];
                orp[i*9 + r*3 + 0] = m0*Ra00 + m1*Ra10 + m2*Ra20;
                orp[i*9 + r*3 + 1] = m0*Ra01 + m1*Ra11 + m2*Ra21;
                orp[i*9 + r*3 + 2] = m0*Ra02 + m1*Ra12 + m2*Ra22;
            }
        }
    }
}

extern "C" void kernel_launch(void* const* d_in, const int* in_sizes, int n_in,
                              void* d_out, int out_size, void* d_ws, size_t ws_size,
                              hipStream_t stream) {
    const int B = in_sizes[9];                       // yaw_u is (B,)
    const float* poses_3d = (const float*)d_in[0];
    const float* rot_mats = (const float*)d_in[1];
    const float* axis_in  = (const float*)d_in[2];
    const float* angle_u  = (const float*)d_in[3];
    const float* scale_u  = (const float*)d_in[4];
    const float* Kin      = (const float*)d_in[5];
    const float* Rin      = (const float*)d_in[6];
    const float* tin      = (const float*)d_in[7];
    const float* focal_u  = (const float*)d_in[8];
    const float* yaw_u    = (const float*)d_in[9];
    const float* pitch_u  = (const float*)d_in[10];
    const float* t_jit    = (const float*)d_in[11];

    float* out = (float*)d_out;
    float* o_poses = out;                            // (B,24,3)
    float* o_2d    = out + (size_t)B * 72;           // (B,24,2)
    float* o_rots  = out + (size_t)B * 120;          // (B,24,3,3)
    float* o_K     = out + (size_t)B * 336;          // (B,3,3)
    float* o_R     = out + (size_t)B * 345;          // (B,3,3)
    float* o_t     = out + (size_t)B * 354;          // (B,3)

    dim3 grid((B + TPB - 1) / TPB), block(TPB);
    size_t shmem = (size_t)WAVES_PER_BLOCK * 2 * TILE_FLOATS * sizeof(float); // 147456 B
    poseaug_kernel<<<grid, block, shmem, stream>>>(
        poses_3d, rot_mats, axis_in, angle_u, scale_u, Kin, Rin, tin,
        focal_u, yaw_u, pitch_u, t_jit,
        o_poses, o_2d, o_rots, o_K, o_R, o_t, B);
}